// MILLoss_15985868275848
// MI455X (gfx1250) — compile-verified
//
#include <hip/hip_runtime.h>
#include <math.h>

// Problem constants from the reference: B=128, H=W=512, 50 zones.
#define BATCH      128
#define HW         (512 * 512)          // pixels per image
#define CHUNKS     64                   // chunks per image
#define CHUNK_PIX  (HW / CHUNKS)        // 4096 pixels per block
#define THREADS    256                  // 8 wave32s per block
#define NEGV       (-1.0e30f)

typedef int   v4i __attribute__((ext_vector_type(4)));
typedef float v4f __attribute__((ext_vector_type(4)));
typedef float v2f __attribute__((ext_vector_type(2)));
typedef float v8f __attribute__((ext_vector_type(8)));

// ---------------------------------------------------------------------------
// Kernel 1: streaming masked-max partial reduction.
// grid = (CHUNKS, BATCH), block = 256. Each thread streams 16 pixels via
// 4x (int4 zone + float4 logit) non-temporal 128-bit loads.
// ---------------------------------------------------------------------------
__global__ void mil_partial_kernel(const float* __restrict__ logits,
                                   const int*   __restrict__ zones,
                                   const int*   __restrict__ cats,
                                   float*       __restrict__ pmax,
                                   int*         __restrict__ pany) {
  const int b   = blockIdx.y;
  const int c   = blockIdx.x;
  const int tid = threadIdx.x;
  const int out_idx = b * CHUNKS + c;

  const int cat = cats[b];            // scalar (SMEM) load, uniform per block
  if (cat <= 0) {
    // mask = (zone==cat) && (zone>0) is provably empty
    if (tid == 0) { pmax[out_idx] = NEGV; pany[out_idx] = 0; }
    return;
  }

  const v4i* __restrict__ zv = (const v4i*)zones;
  const v4f* __restrict__ lv = (const v4f*)logits;
  const size_t vbase = (size_t)b * (HW / 4) + (size_t)c * (CHUNK_PIX / 4);

  float m   = NEGV;
  int   any = 0;
#pragma unroll
  for (int i = 0; i < 4; ++i) {
    const size_t idx = vbase + (size_t)i * THREADS + tid;
    v4i z = __builtin_nontemporal_load(&zv[idx]);   // NT: don't pollute L2
    v4f l = __builtin_nontemporal_load(&lv[idx]);
    if (z.x == cat) { any = 1; m = fmaxf(m, l.x); } // zone==cat && cat>0 => zone>0
    if (z.y == cat) { any = 1; m = fmaxf(m, l.y); }
    if (z.z == cat) { any = 1; m = fmaxf(m, l.z); }
    if (z.w == cat) { any = 1; m = fmaxf(m, l.w); }
  }

  __shared__ float smax[THREADS];
  __shared__ int   sany[THREADS];
  smax[tid] = m;
  sany[tid] = any;
  __syncthreads();
#pragma unroll
  for (int s = THREADS / 2; s > 0; s >>= 1) {
    if (tid < s) {
      smax[tid] = fmaxf(smax[tid], smax[tid + s]);
      sany[tid] |= sany[tid + s];
    }
    __syncthreads();
  }
  if (tid == 0) { pmax[out_idx] = smax[0]; pany[out_idx] = sany[0]; }
}

// ---------------------------------------------------------------------------
// Kernel 2: per-batch final max + BCE-with-logits term.
// grid = BATCH, block = CHUNKS (64).
// ---------------------------------------------------------------------------
__global__ void mil_loss_kernel(const float* __restrict__ pmax,
                                const int*   __restrict__ pany,
                                const int*   __restrict__ labels,
                                float*       __restrict__ loss_terms) {
  const int b   = blockIdx.x;
  const int tid = threadIdx.x;   // 0..63

  __shared__ float smax[CHUNKS];
  __shared__ int   sany[CHUNKS];
  smax[tid] = pmax[b * CHUNKS + tid];
  sany[tid] = pany[b * CHUNKS + tid];
  __syncthreads();
#pragma unroll
  for (int s = CHUNKS / 2; s > 0; s >>= 1) {
    if (tid < s) {
      smax[tid] = fmaxf(smax[tid], smax[tid + s]);
      sany[tid] |= sany[tid + s];
    }
    __syncthreads();
  }
  if (tid == 0) {
    const float sc = sany[0] ? smax[0] : 0.0f;
    const float y  = (float)labels[b];
    // max(s,0) - s*y + log1p(exp(-|s|))
    loss_terms[b] = fmaxf(sc, 0.0f) - sc * y + log1pf(expf(-fabsf(sc)));
  }
}

// ---------------------------------------------------------------------------
// Kernel 3: sum 128 loss terms with V_WMMA_F32_16X16X4_F32 (B = ones).
// D[m][n] = sum_k A[m][k] * 1 = rowsum(A) -> every column of D is identical.
// A f32 16x4 layout: lanes 0-15 hold (K=0,K=1) of row M=lane, lanes 16-31
// hold (K=2,K=3) of row M=lane-16. Two accumulating WMMAs cover 128 terms.
// After WMMA, lane n (n<16) holds rowsums 0..7 in c[0..7]; lanes 16-31 hold
// rowsums 8..15 -> lane-local sum + shfl_xor(16) gives the grand total.
// 1 block of 32 threads, no divergence -> EXEC all ones as WMMA requires.
// ---------------------------------------------------------------------------
__global__ void mil_final_wmma_kernel(const float* __restrict__ terms,
                                      float*       __restrict__ out) {
  const int lane = threadIdx.x;                 // 0..31, wave32
  v2f bmat; bmat.x = 1.0f; bmat.y = 1.0f;       // B = all ones (4x16)
  v8f c = {};                                   // C starts at zero
#pragma unroll
  for (int p = 0; p < 2; ++p) {
    const int idx = p * 64 + (lane & 15) * 4 + (lane >> 4) * 2;
    v2f a; a.x = terms[idx]; a.y = terms[idx + 1];
    // 8 args: (neg_a, A, neg_b, B, c_mod, C, reuse_a, reuse_b)
    c = __builtin_amdgcn_wmma_f32_16x16x4_f32(false, a, false, bmat,
                                              (short)0, c, false, false);
  }
  float partial = 0.0f;
#pragma unroll
  for (int j = 0; j < 8; ++j) partial += c[j];
  const float total = partial + __shfl_xor(partial, 16, 32);
  if (lane == 0) out[0] = total * (1.0f / (float)BATCH);
}

// ---------------------------------------------------------------------------
// Launcher. Inputs: pixel_logits f32[128*512*512], zone_patches i32[same],
// cats i32[128], labels i32[128]. Output: scalar f32 loss.
// Workspace: pmax f32[128*64] | pany i32[128*64] | terms f32[128] (~66 KB).
// ---------------------------------------------------------------------------
extern "C" void kernel_launch(void* const* d_in, const int* in_sizes, int n_in,
                              void* d_out, int out_size, void* d_ws, size_t ws_size,
                              hipStream_t stream) {
  const float* logits = (const float*)d_in[0];
  const int*   zones  = (const int*)d_in[1];
  const int*   cats   = (const int*)d_in[2];
  const int*   labels = (const int*)d_in[3];
  float*       out    = (float*)d_out;

  char* ws = (char*)d_ws;
  float* pmax  = (float*)ws;
  int*   pany  = (int*)(ws + (size_t)BATCH * CHUNKS * sizeof(float));
  float* terms = (float*)(ws + (size_t)2 * BATCH * CHUNKS * sizeof(float));

  mil_partial_kernel<<<dim3(CHUNKS, BATCH), THREADS, 0, stream>>>(
      logits, zones, cats, pmax, pany);
  mil_loss_kernel<<<BATCH, CHUNKS, 0, stream>>>(pmax, pany, labels, terms);
  mil_final_wmma_kernel<<<1, 32, 0, stream>>>(terms, out);
}